// RelationalGraphConvolution_7541962571802
// MI455X (gfx1250) — compile-verified
//
#include <hip/hip_runtime.h>
#include <hip/hip_bf16.h>

#define E_EDGES   131072
#define IN_F      128
#define OUT_F     128
#define NUM_REL   16
#define NUM_BASES 4

typedef __attribute__((ext_vector_type(2))) float v2f;
typedef __attribute__((ext_vector_type(8))) float v8f;

// ---------------------------------------------------------------------------
// Kernel 1: tiny prep. v[r][i] = sum_b att[r,b] * (sum_o basis[b,i,o]);
// also zero the scalar accumulator. One block of 128 threads.
// ---------------------------------------------------------------------------
__global__ __launch_bounds__(128) void rgc_prep(const float* __restrict__ basis,
                                                const float* __restrict__ att,
                                                float* __restrict__ vrel,
                                                float* __restrict__ acc) {
    const int i = threadIdx.x;             // feature index 0..127
    float bsum[NUM_BASES];
#pragma unroll
    for (int b = 0; b < NUM_BASES; ++b) {
        const float* row = basis + ((size_t)b * IN_F + i) * OUT_F;
        float s = 0.f;
        for (int o = 0; o < OUT_F; o += 4) {
            s += row[o] + row[o + 1] + row[o + 2] + row[o + 3];
        }
        bsum[b] = s;
    }
#pragma unroll
    for (int r = 0; r < NUM_REL; ++r) {
        float s = 0.f;
#pragma unroll
        for (int b = 0; b < NUM_BASES; ++b) s += att[r * NUM_BASES + b] * bsum[b];
        vrel[r * IN_F + i] = s;
    }
    if (i == 0) *acc = 0.f;
}

// ---------------------------------------------------------------------------
// Kernel 2: scalar reduction  S_raw = sum_e norm[e] * dot(x[e], v[edge_type[e]])
// One wave32 per edge; coalesced x reads; shuffle reduce; one global atomic
// per block (8 waves/block -> LDS partial first).
// ---------------------------------------------------------------------------
__global__ __launch_bounds__(256) void rgc_reduce(const float* __restrict__ x,
                                                  const int* __restrict__ edge_type,
                                                  const float* __restrict__ edge_norm,
                                                  const float* __restrict__ vrel,
                                                  float* __restrict__ acc) {
    __shared__ float partial;
    if (threadIdx.x == 0) partial = 0.f;
    __syncthreads();

    const int lane = threadIdx.x & 31;
    const int wv   = threadIdx.x >> 5;            // wave in block, 0..7
    const int e    = blockIdx.x * 8 + wv;

    float s = 0.f;
    if (e < E_EDGES) {
        const int r = edge_type[e];
        const float* xr = x + (size_t)e * IN_F;
        const float* vr = vrel + (size_t)r * IN_F;
#pragma unroll
        for (int i = lane; i < IN_F; i += 32) s += xr[i] * vr[i];
        s *= edge_norm[e];                        // uniform per wave; sum == norm*dot
    }
#pragma unroll
    for (int off = 16; off > 0; off >>= 1) s += __shfl_down(s, off, 32);
    if (lane == 0) atomicAdd(&partial, s);
    __syncthreads();
    if (threadIdx.x == 0) atomicAdd(acc, partial);
}

// ---------------------------------------------------------------------------
// Kernel 3: out = S + x @ weight + bias, fp32 WMMA 16x16x4.
// Block = 256 threads = 8 waves; block owns a 16-row stripe, wave w owns
// columns [16w, 16w+16). weight (64 KB) staged in LDS once per block.
//
// Fragment layouts per CDNA5 ISA 7.12.2 (wave32):
//   A (16x4 f32):  lane L holds row M=L%16; VGPR0/1 = K-pair {0,1} (L<16)
//                  or {2,3} (L>=16)  -> a = x[row][k+kk .. k+kk+1], kk=(L/16)*2
//   B (4x16 f32):  row K striped across lanes within a VGPR (mirror of A)
//   C/D (16x16):   VGPR v, lane L -> M = v + 8*(L/16), N = L%16
// ---------------------------------------------------------------------------
__global__ __launch_bounds__(256) void rgc_gemm(const float* __restrict__ x,
                                                const float* __restrict__ weight,
                                                const float* __restrict__ bias,
                                                const float* __restrict__ ws,
                                                float* __restrict__ out) {
    __shared__ float wlds[IN_F * OUT_F];          // 64 KB of the 320 KB WGP LDS

    const int tid = threadIdx.x;
    // Stage weight into LDS with 128-bit copies.
    {
        const float4* src = (const float4*)weight;
        float4* dst = (float4*)wlds;
#pragma unroll
        for (int i = tid; i < (IN_F * OUT_F) / 4; i += 256) dst[i] = src[i];
    }
    __syncthreads();

    const int lane = tid & 31;
    const int wv   = tid >> 5;                    // 0..7 -> column tile
    const int m0   = blockIdx.x * 16;
    const int n0   = wv * 16;
    const int mrow = lane & 15;                   // A-fragment row
    const int n    = lane & 15;                   // B/C/D-fragment column
    const int kk   = (lane >> 4) * 2;             // K-pair selector

    const float* xrow = x + (size_t)(m0 + mrow) * IN_F;

    v8f acc = {};
#pragma unroll
    for (int k = 0; k < IN_F; k += 4) {
        v2f a;
        a.x = xrow[k + kk];
        a.y = xrow[k + kk + 1];
        v2f b;
        b.x = wlds[(k + kk) * OUT_F + n0 + n];
        b.y = wlds[(k + kk + 1) * OUT_F + n0 + n];
        acc = __builtin_amdgcn_wmma_f32_16x16x4_f32(
            /*neg_a=*/false, a, /*neg_b=*/false, b,
            /*c_mod=*/(short)0, acc, /*reuse_a=*/false, /*reuse_b=*/false);
    }

    const float S = ws[0] * (1.0f / ((float)E_EDGES * (float)OUT_F));
    const float bn = bias[n0 + n] + S;
    const int rbase = m0 + (lane >> 4) * 8;
#pragma unroll
    for (int v = 0; v < 8; ++v) {
        out[(size_t)(rbase + v) * OUT_F + n0 + n] = acc[v] + bn;
    }
}

// ---------------------------------------------------------------------------
// Launcher. Inputs (setup_inputs order):
//   0:x  1:edge_index(unused by reference!)  2:edge_type  3:edge_norm
//   4:basis  5:att  6:weight  7:bias
// Workspace layout (floats): [0] scalar accumulator, [32..32+2048) vrel[16][128]
// ---------------------------------------------------------------------------
extern "C" void kernel_launch(void* const* d_in, const int* in_sizes, int n_in,
                              void* d_out, int out_size, void* d_ws, size_t ws_size,
                              hipStream_t stream) {
    const float* x         = (const float*)d_in[0];
    const int*   edge_type = (const int*)d_in[2];
    const float* edge_norm = (const float*)d_in[3];
    const float* basis     = (const float*)d_in[4];
    const float* att       = (const float*)d_in[5];
    const float* weight    = (const float*)d_in[6];
    const float* bias      = (const float*)d_in[7];
    float*       out       = (float*)d_out;

    float* wsf  = (float*)d_ws;
    float* acc  = wsf;            // 1 float
    float* vrel = wsf + 32;       // 16*128 floats, 128B-aligned

    rgc_prep<<<1, 128, 0, stream>>>(basis, att, vrel, acc);
    rgc_reduce<<<E_EDGES / 8, 256, 0, stream>>>(x, edge_type, edge_norm, vrel, acc);
    rgc_gemm<<<E_EDGES / 16, 256, 0, stream>>>(x, weight, bias, acc, out);
}